// FNO_47502338294499
// MI455X (gfx1250) — compile-verified
//
#include <hip/hip_runtime.h>
#include <math.h>

// ---------------------------------------------------------------------------
// FNO layer collapsed to a per-batch spectral filter:
//   out[b,s] = E*x[b,s] + dec_b + (1/S) * sum_k Cf[b,k] * T[k,s]
// with Cf built from Xhat[b, 0..63] = 64-bin DFT of x[b,:] and
//   g[f]  = sum_{n,d} enc_w[n] * W[n,d,f] * dec_w[d]   (complex)
//   E     = enc_w . dec_w
//   C0r   = S * sum_{n,d} enc_b[n] * w_r[n,d,0] * dec_w[d]
// Both DFT directions are F32 WMMA GEMMs (V_WMMA_F32_16X16X4_F32).
// Forward GEMM is K-split 32 ways (512 waves) for occupancy; partials are
// folded deterministically in fno_make_cf.
// ---------------------------------------------------------------------------

typedef __attribute__((ext_vector_type(2))) float v2f;
typedef __attribute__((ext_vector_type(8))) float v8f;

#define BATCH 32
#define SLEN  8192
#define MODES 64
#define K2    128     // 2*MODES (cos rows then sin rows)
#define KCH   32      // forward K-split chunks
#define KLEN  (SLEN / KCH)   // 256 samples per chunk

// workspace offsets (in floats); total ~8.55 MB
#define T_OFF   0u          // T [128][8192]: rows 0..63 cos(2pi f s/S), 64..127 sin
#define TT_OFF  1048576u    // Tt[8192][128]: transpose of T (forward-GEMM B)
#define GR_OFF  2097152u    // g_r[64]
#define GI_OFF  2097216u    // g_i[64]
#define SC_OFF  2097280u    // [0]=E, [1]=C0r
#define CF_OFF  2097344u    // Cf[32][128]
#define XP_OFF  2101440u    // XP[32 chunks][32 b][128 j] partial DFT sums

// --- build trig tables (both orientations) ---------------------------------
__global__ void fno_fill_tables(float* __restrict__ ws) {
    unsigned tid = blockIdx.x * blockDim.x + threadIdx.x;   // 0..1048575
    unsigned j = tid >> 13;                                  // 0..127
    unsigned s = tid & 8191u;
    unsigned ph = ((j & 63u) * s) & 8191u;                   // exact mod-S phase
    float ang = 7.66990393942820595e-4f * (float)ph;         // 2*pi/8192
    float val = (j < 64u) ? cosf(ang) : sinf(ang);
    ws[T_OFF  + j * 8192u + s] = val;
    ws[TT_OFF + s * 128u  + j] = val;
}

// --- fold weights: g[f], E, C0r (LDS tree reduction) -----------------------
__global__ void fno_precompute_g(const float* __restrict__ enc_w,
                                 const float* __restrict__ enc_b,
                                 const float* __restrict__ w_r,
                                 const float* __restrict__ w_i,
                                 const float* __restrict__ dec_w,
                                 float* __restrict__ ws) {
    __shared__ float red0[128], red1[128], red2[128];
    int m = blockIdx.x;          // mode 0..63
    int t = threadIdx.x;         // 0..127
    float sr = 0.f, si = 0.f, c0 = 0.f;
    for (int p = t; p < 4096; p += 128) {     // p = n*64 + d
        int n = p >> 6, d = p & 63;
        float dw = dec_w[d];
        float wt = enc_w[n] * dw;
        float wr = w_r[p * 64 + m];
        float wi = w_i[p * 64 + m];
        sr += wt * wr;
        si += wt * wi;
        c0 += enc_b[n] * dw * wr;             // used only when m==0
    }
    red0[t] = sr; red1[t] = si; red2[t] = c0;
    __syncthreads();
    for (int off = 64; off > 0; off >>= 1) {
        if (t < off) {
            red0[t] += red0[t + off];
            red1[t] += red1[t + off];
            red2[t] += red2[t + off];
        }
        __syncthreads();
    }
    if (t == 0) {
        ws[GR_OFF + m] = red0[0];
        ws[GI_OFF + m] = red1[0];
        if (m == 0) {
            ws[SC_OFF + 1] = 8192.0f * red2[0];   // C0r
            float e = 0.f;
            for (int n = 0; n < 64; ++n) e += enc_w[n] * dec_w[n];
            ws[SC_OFF + 0] = e;                   // E
        }
    }
}

// --- forward 64-bin DFT partials: XP[c] += x[32][Kc] @ Tt[Kc][128] ---------
// one wave per (16x16 tile, K-chunk); 16 tiles x 32 chunks = 512 waves
__global__ void fno_fwd_dft(const float* __restrict__ x, float* __restrict__ ws) {
    int wave = (blockIdx.x * blockDim.x + threadIdx.x) >> 5;  // 0..511
    int lane = threadIdx.x & 31;
    int lo = lane & 15, hi = lane >> 4;
    int tile  = wave & 15;
    int chunk = wave >> 4;            // 0..31
    int m0 = (tile >> 3) * 16;        // batch tile: 0 or 16
    int j0 = (tile & 7) * 16;         // freq-pair tile: 0..112
    int k0 = chunk * KLEN;
    const float* xrow = x + (unsigned)(m0 + lo) * SLEN;   // A: row M=lo
    const float* tt = ws + TT_OFF;
    v8f c = {};
    for (int k = k0; k < k0 + KLEN; k += 4) {
        int ka = k + 2 * hi;          // A VGPR0: K{0,2}, VGPR1: K{1,3}
        v2f a; a.x = xrow[ka]; a.y = xrow[ka + 1];
        v2f b;
        b.x = tt[(unsigned)ka * 128u + (unsigned)(j0 + lo)];
        b.y = tt[(unsigned)(ka + 1) * 128u + (unsigned)(j0 + lo)];
        c = __builtin_amdgcn_wmma_f32_16x16x4_f32(false, a, false, b,
                                                  (short)0, c, false, false);
    }
    float* XP = ws + XP_OFF + (unsigned)chunk * 4096u;
#pragma unroll
    for (int v = 0; v < 8; ++v)       // D: row M = v + 8*hi, col N = lo
        XP[(unsigned)(m0 + v + 8 * hi) * 128u + (unsigned)(j0 + lo)] = c[v];
}

// --- fold K-chunk partials + spectral filter coefficients Cf[32][128] ------
__global__ void fno_make_cf(float* __restrict__ ws) {
    int tid = blockIdx.x * blockDim.x + threadIdx.x;   // 0..2047
    int b = tid >> 6, f = tid & 63;
    float Xc = 0.f, Xs = 0.f;
#pragma unroll 4
    for (int cidx = 0; cidx < KCH; ++cidx) {           // fixed order: deterministic
        const float* xp = ws + XP_OFF + (unsigned)cidx * 4096u + (unsigned)b * 128u;
        Xc += xp[f];
        Xs += xp[64 + f];
    }
    float E   = ws[SC_OFF + 0];
    float C0r = ws[SC_OFF + 1];
    float hr = ws[GR_OFF + f] - E;
    float hi = ws[GI_OFF + f];
    float cc, sc;
    if (f == 0) {                      // DC: irfft ignores imag of bin 0
        cc = hr * Xc + C0r;
        sc = 0.0f;
    } else {                           // Re(2*(h*Xhat)*e^{i theta})
        cc =  2.0f * (hr * Xc + hi * Xs);
        sc = -2.0f * (hi * Xc - hr * Xs);
    }
    ws[CF_OFF + b * 128 + f]      = cc;
    ws[CF_OFF + b * 128 + 64 + f] = sc;
}

// --- inverse: out = E*x + dec_b + (1/S) * Cf[32][128] @ T[128][8192] -------
// one wave per 16x16 output tile; 1024 waves; K=128 fully unrolled
__global__ void fno_inv_dft(const float* __restrict__ x,
                            const float* __restrict__ dec_b,
                            const float* __restrict__ ws,
                            float* __restrict__ out) {
    int wave = (blockIdx.x * blockDim.x + threadIdx.x) >> 5;  // 0..1023
    int lane = threadIdx.x & 31;
    int lo = lane & 15, hi = lane >> 4;
    int m0 = (wave >> 9) * 16;         // batch tile
    int s0 = (wave & 511) * 16;        // sample tile
    // warm the epilogue x tile (global_prefetch_b8) while WMMAs run
    __builtin_prefetch(&x[(unsigned)(m0 + 8 * hi) * SLEN + (unsigned)(s0 + lo)], 0, 1);
    const float* cf = ws + CF_OFF + (unsigned)(m0 + lo) * 128u;
    const float* T  = ws + T_OFF;
    v8f c = {};
#pragma unroll
    for (int k = 0; k < K2; k += 4) {
        int ka = k + 2 * hi;
        v2f a; a.x = cf[ka]; a.y = cf[ka + 1];
        v2f b;
        b.x = T[(unsigned)ka * 8192u + (unsigned)(s0 + lo)];
        b.y = T[(unsigned)(ka + 1) * 8192u + (unsigned)(s0 + lo)];
        c = __builtin_amdgcn_wmma_f32_16x16x4_f32(false, a, false, b,
                                                  (short)0, c, false, false);
    }
    float E  = ws[SC_OFF + 0];
    float db = dec_b[0];
    const float inv_s = 1.0f / 8192.0f;
#pragma unroll
    for (int v = 0; v < 8; ++v) {
        unsigned bb = (unsigned)(m0 + v + 8 * hi);
        unsigned ss = (unsigned)(s0 + lo);
        out[bb * SLEN + ss] = E * x[bb * SLEN + ss] + db + c[v] * inv_s;
    }
}

extern "C" void kernel_launch(void* const* d_in, const int* in_sizes, int n_in,
                              void* d_out, int out_size, void* d_ws, size_t ws_size,
                              hipStream_t stream) {
    const float* x     = (const float*)d_in[0];   // [32,8192,1]
    const float* enc_w = (const float*)d_in[1];   // [64,1]
    const float* enc_b = (const float*)d_in[2];   // [64]
    const float* w_r   = (const float*)d_in[3];   // [64,64,64]
    const float* w_i   = (const float*)d_in[4];   // [64,64,64]
    const float* dec_w = (const float*)d_in[5];   // [1,64]
    const float* dec_b = (const float*)d_in[6];   // [1]
    float* ws  = (float*)d_ws;                    // needs ~8.6 MB
    float* out = (float*)d_out;                   // [32,8192,1] f32

    fno_fill_tables<<<4096, 256, 0, stream>>>(ws);
    fno_precompute_g<<<64, 128, 0, stream>>>(enc_w, enc_b, w_r, w_i, dec_w, ws);
    fno_fwd_dft<<<128, 128, 0, stream>>>(x, ws);     // 512 waves, WMMA f32
    fno_make_cf<<<8, 256, 0, stream>>>(ws);
    fno_inv_dft<<<256, 128, 0, stream>>>(x, dec_b, ws, out);  // 1024 waves, WMMA f32
}